// Self_Attn_81346680587159
// MI455X (gfx1250) — compile-verified
//
#include <hip/hip_runtime.h>
#include <hip/hip_bf16.h>

// Problem constants (reference: B=4, C=128, H=W=64)
#define BB 4
#define CC 128
#define NN 4096          // H*W
#define DD 16            // DQK
#define EPITCH 4100      // 4096 + 4 pad: avoids LDS bank conflicts on column reads

typedef __attribute__((ext_vector_type(2))) float v2f;
typedef __attribute__((ext_vector_type(8))) float v8f;

// -------------------------------------------------------------------------
// Kernel 1: 1x1-conv projections.  x:(B,C,N) ->
//   q   : (B, N, 16)            row-major, feeds WMMA A fragments
//   kT2 : (B, 8, N, 2)          K-rows pair-interleaved: (d,d+1) adjacent
//                               -> phase-A B-fragment = one b64 load
//   vT2 : (B, N/2, C, 2)        V rows pair-interleaved: (n,n+1) adjacent
//                               -> phase-C B-fragment = one b64 load
// -------------------------------------------------------------------------
__global__ void proj_kernel(const float* __restrict__ x,
                            const float* __restrict__ Wq, const float* __restrict__ bq,
                            const float* __restrict__ Wk, const float* __restrict__ bk,
                            const float* __restrict__ Wv, const float* __restrict__ bv,
                            float* __restrict__ q, float* __restrict__ kT2,
                            float* __restrict__ vT2)
{
    __shared__ float xs[CC][33];                 // 33: bank-conflict pad
    const int b   = blockIdx.x / (NN / 32);
    const int n0  = (blockIdx.x % (NN / 32)) * 32;
    const int tid = threadIdx.x;                 // 256 threads

    for (int idx = tid; idx < CC * 32; idx += 256) {
        int c = idx >> 5, j = idx & 31;
        xs[c][j] = x[((long)b * CC + c) * NN + n0 + j];
    }
    __syncthreads();

    // 160 output channels (16 q + 16 k + 128 v) x 32 pixels
    for (int t = tid; t < 160 * 32; t += 256) {
        int j = t & 31;
        int o = t >> 5;                          // uniform per wave32
        const float* Wrow;
        float s;
        if (o < 16)      { Wrow = Wq + o * CC;        s = bq[o]; }
        else if (o < 32) { Wrow = Wk + (o - 16) * CC; s = bk[o - 16]; }
        else             { Wrow = Wv + (o - 32) * CC; s = bv[o - 32]; }
        #pragma unroll 8
        for (int c = 0; c < CC; ++c) s += Wrow[c] * xs[c][j];
        const int n = n0 + j;
        if (o < 16) {
            q[((long)b * NN + n) * DD + o] = s;
        } else if (o < 32) {
            const int d = o - 16;
            kT2[(((long)b * 8 + (d >> 1)) * NN + n) * 2 + (d & 1)] = s;
        } else {
            const int c = o - 32;
            vT2[(((long)b * (NN / 2) + (n >> 1)) * CC + c) * 2 + (n & 1)] = s;
        }
    }
}

// -------------------------------------------------------------------------
// Kernel 2: fused attention.  One workgroup (256 thr = 8 wave32) owns one
// (batch, 16-row) block.  Energy row-block lives entirely in LDS (16x4096
// fp32 = 256KB, legal only on CDNA5's 320KB/WG LDS).
//   Phase A: E = Qblk(16x16) x K(16xN)   via V_WMMA_F32_16X16X4_F32
//   Phase B: row softmax in LDS; attention rows -> HBM once (NT stores,
//            keeps the 192MB L2 for the reused kT/vT/x working set)
//   Phase C: out(16x128) = P(16xN) x Vt(Nx128) via WMMA
//   Epilogue: stage out-tile in LDS, coalesced b128 gamma*out + x
// -------------------------------------------------------------------------
__global__ void attn_kernel(const float* __restrict__ q,
                            const float* __restrict__ kT2,
                            const float* __restrict__ vT2,
                            const float* __restrict__ x,
                            const float* __restrict__ gamma,
                            float* __restrict__ outp,
                            float* __restrict__ attn)
{
    extern __shared__ float smem[];
    float* E   = smem;                    // 16 * EPITCH floats
    float* Qs  = smem + 16 * EPITCH;      // 256 floats
    float* red = Qs + 256;                // 256 floats

    const int b    = blockIdx.x >> 8;     // 256 row-blocks per batch
    const int m0   = (blockIdx.x & 255) * 16;
    const int tid  = threadIdx.x;
    const int lane = tid & 31;
    const int wave = tid >> 5;
    const int mL   = lane & 15;           // A-row / B-col / D-col lane index
    const int half = lane >> 4;           // 0: lanes 0-15, 1: lanes 16-31
    const int koff = half * 2;            // ISA 7.12.2: upper half holds K+2

    // Stage Q block (16 rows x 16 d), build the 4 A-fragments (K chunks of 4)
    Qs[tid] = q[((long)b * NN + m0 + (tid >> 4)) * DD + (tid & 15)];
    __syncthreads();

    v2f afr[4];
    #pragma unroll
    for (int kc = 0; kc < 4; ++kc) {
        afr[kc].x = Qs[mL * DD + kc * 4 + koff];
        afr[kc].y = Qs[mL * DD + kc * 4 + koff + 1];
    }

    // ---- Phase A: energy block into LDS -------------------------------
    const float* kb = kT2 + (long)b * DD * NN;   // (8, N, 2) pair-interleaved
    for (int nt = wave * 32; nt < wave * 32 + 32; ++nt) {
        const int n0t = nt * 16;
        v8f acc = {};
        #pragma unroll
        for (int kc = 0; kc < 4; ++kc) {
            // K-row pair index: (kc*4 + koff)/2 = kc*2 + half
            const v2f bfr = *(const v2f*)&kb[((long)(kc * 2 + half) * NN + n0t + mL) * 2];
            acc = __builtin_amdgcn_wmma_f32_16x16x4_f32(
                false, afr[kc], false, bfr, (short)0, acc, false, false);
        }
        #pragma unroll
        for (int i = 0; i < 8; ++i)
            E[(i + 8 * half) * EPITCH + n0t + mL] = acc[i];
    }
    __syncthreads();

    // ---- Phase B: softmax over each of the 16 rows (16 threads / row) --
    {
        const int r = tid >> 4, j = tid & 15;
        float* Erow = E + r * EPITCH;

        float lmax = -3.4e38f;
        for (int n = j; n < NN; n += 16) lmax = fmaxf(lmax, Erow[n]);
        red[tid] = lmax;
        __syncthreads();
        float rmax = red[r * 16];
        #pragma unroll
        for (int t2 = 1; t2 < 16; ++t2) rmax = fmaxf(rmax, red[r * 16 + t2]);
        __syncthreads();                       // red reused below

        float lsum = 0.f;
        for (int n = j; n < NN; n += 16) {
            float e = __expf(Erow[n] - rmax);
            Erow[n] = e;
            lsum += e;
        }
        red[tid] = lsum;
        __syncthreads();
        float rsum = 0.f;
        #pragma unroll
        for (int t2 = 0; t2 < 16; ++t2) rsum += red[r * 16 + t2];
        const float inv = 1.0f / rsum;

        float* arow = attn + ((long)b * NN + m0 + r) * (long)NN;
        for (int n = j; n < NN; n += 16) {
            float p = Erow[n] * inv;
            Erow[n] = p;                        // normalized P stays in LDS
            __builtin_nontemporal_store(p, arow + n);  // write-once, bypass L2 reuse
        }
    }
    __syncthreads();

    // ---- Phase C: out block = P x Vt, each wave owns a 16-wide C tile --
    const int c0 = wave * 16;
    v8f acc = {};
    {
        const float* vb = vT2 + (long)b * NN * CC;   // (N/2, C, 2) interleaved
        #pragma unroll 8
        for (int nc = 0; nc < NN / 4; ++nc) {
            const int n = nc * 4 + koff;             // even
            // A fragment: adjacent words in LDS -> one ds_load_b64
            const v2f a2 = *(const v2f*)&E[mL * EPITCH + n];
            // B fragment: row pair (n, n+1) interleaved -> one global b64
            const v2f bfr = *(const v2f*)&vb[((long)(nc * 2 + half) * CC + c0 + mL) * 2];
            acc = __builtin_amdgcn_wmma_f32_16x16x4_f32(
                false, a2, false, bfr, (short)0, acc, false, false);
        }
    }

    // ---- Epilogue: stage 16x128 out-tile in LDS, coalesced residual ----
    __syncthreads();                       // E is dead; safe to reuse smem
    float* Osh = smem;                     // [c][m] with pitch 17 (8.5 KB)
    #pragma unroll
    for (int i = 0; i < 8; ++i)
        Osh[(c0 + mL) * 17 + (i + 8 * half)] = acc[i];
    __syncthreads();
    {
        const int c  = tid >> 1;                 // 128 channels / 2 threads
        const int h8 = (tid & 1) * 8;            // 8 contiguous m per thread
        const float g = gamma[0];
        const long base = ((long)b * CC + c) * NN + m0 + h8;
        const float4 x0 = *(const float4*)(x + base);
        const float4 x1 = *(const float4*)(x + base + 4);
        const float* Oc = Osh + c * 17 + h8;
        float4 o0, o1;
        o0.x = g * Oc[0] + x0.x;  o0.y = g * Oc[1] + x0.y;
        o0.z = g * Oc[2] + x0.z;  o0.w = g * Oc[3] + x0.w;
        o1.x = g * Oc[4] + x1.x;  o1.y = g * Oc[5] + x1.y;
        o1.z = g * Oc[6] + x1.z;  o1.w = g * Oc[7] + x1.w;
        *(float4*)(outp + base)     = o0;
        *(float4*)(outp + base + 4) = o1;
    }
}

// -------------------------------------------------------------------------
extern "C" void kernel_launch(void* const* d_in, const int* in_sizes, int n_in,
                              void* d_out, int out_size, void* d_ws, size_t ws_size,
                              hipStream_t stream)
{
    const float* x     = (const float*)d_in[0];
    const float* Wq    = (const float*)d_in[1];
    const float* bq    = (const float*)d_in[2];
    const float* Wk    = (const float*)d_in[3];
    const float* bk    = (const float*)d_in[4];
    const float* Wv    = (const float*)d_in[5];
    const float* bv    = (const float*)d_in[6];
    const float* gamma = (const float*)d_in[7];

    float* outp = (float*)d_out;                       // (B,C,N)
    float* attn = outp + (long)BB * CC * NN;           // (B,N,N)

    // workspace: q (B,N,16) | kT2 (B,8,N,2) | vT2 (B,N/2,128,2)  -> ~10.5 MB
    float* q   = (float*)d_ws;
    float* kT2 = q   + (long)BB * NN * DD;
    float* vT2 = kT2 + (long)BB * DD * NN;

    proj_kernel<<<dim3(BB * (NN / 32)), 256, 0, stream>>>(
        x, Wq, bq, Wk, bk, Wv, bv, q, kT2, vT2);

    const size_t smem = (16 * EPITCH + 256 + 256) * sizeof(float); // ~264.5 KB
    attn_kernel<<<dim3(BB * (NN / 16)), 256, smem, stream>>>(
        q, kT2, vT2, x, gamma, outp, attn);
}